// myNet_91319594647825
// MI455X (gfx1250) — compile-verified
//
#include <hip/hip_runtime.h>
#include <cstddef>

// ---------------------------------------------------------------------------
// Types for CDNA5 WMMA f32 16x16x4  (D = A(16x4) * B(4x16) + C(16x16), wave32)
// ---------------------------------------------------------------------------
typedef __attribute__((ext_vector_type(2))) float v2f;
typedef __attribute__((ext_vector_type(8))) float v8f;

__device__ inline v8f v8f_zero() {
    v8f z;
    for (int i = 0; i < 8; ++i) z[i] = 0.f;
    return z;
}

// One 16x16 output tile of D = A * B, K stepped by 4 via V_WMMA_F32_16X16X4_F32.
// A: row-major [.., lda] (A[m,k] = A[m*lda+k]); B: row-major [.., ldb] (B[k,n]).
// ISA layout (CDNA5 ISA):
//   A 16x4: lanes 0-15 hold M=lane, VGPR0/1 = K0/K1; lanes 16-31 = K2/K3.
//   B 4x16: lanes 0-15 hold N=lane, VGPR0/1 = rows K0/K1; lanes 16-31 = K2/K3.
__device__ inline v8f wmma_f32_mm(const float* __restrict__ A, int lda,
                                  const float* __restrict__ B, int ldb,
                                  int K, int mi, int ni, int lane, v8f acc) {
    const int r    = lane & 15;
    const int koff = (lane >> 4) << 1;  // 0 for lanes 0-15, 2 for lanes 16-31
    for (int k0 = 0; k0 < K; k0 += 4) {
        v2f a, b;
        a[0] = A[(mi * 16 + r) * lda + k0 + koff];
        a[1] = A[(mi * 16 + r) * lda + k0 + koff + 1];
        b[0] = B[(k0 + koff) * ldb + ni * 16 + r];
        b[1] = B[(k0 + koff + 1) * ldb + ni * 16 + r];
        acc = __builtin_amdgcn_wmma_f32_16x16x4_f32(false, a, false, b,
                                                    (short)0, acc, false, false);
    }
    return acc;
}

// C/D 16x16 f32 layout: VGPR r -> M=r (lanes 0-15) / M=8+r (lanes 16-31), N=lane%16.
__device__ inline void wmma_store(float* __restrict__ D, int ldd, int mi, int ni,
                                  int lane, const v8f& acc, float scale) {
    const int col   = ni * 16 + (lane & 15);
    const int rbase = mi * 16 + (lane >> 4) * 8;
    for (int r2 = 0; r2 < 8; ++r2) {
        D[(rbase + r2) * ldd + col] = acc[r2] * scale;
    }
}

// ---------------------------------------------------------------------------
// conv1x1: out[b,o,p] = sum_c w[o,c] * f(in[b,c,p])   (+ optional residuals)
// MODE 0: f = identity
// MODE 1: f = relu(bn(v))                (mu/rs/gamma/beta per channel)
// MODE 2: f = relu(bn(v)) + res[b,c,p]
// outAdd != null: out += outAdd (same layout as out)
// ---------------------------------------------------------------------------
template <int MODE>
__global__ void conv1x1_kernel(const float* __restrict__ in, const float* __restrict__ wgt,
                               float* __restrict__ out,
                               const float* __restrict__ mu, const float* __restrict__ rs,
                               const float* __restrict__ gamma, const float* __restrict__ beta,
                               const float* __restrict__ res,
                               const float* __restrict__ outAdd,
                               int B, int Cin, int Cout, int P) {
    int idx = blockIdx.x * blockDim.x + threadIdx.x;
    int total = B * Cout * P;
    if (idx >= total) return;
    int p = idx % P;
    int o = (idx / P) % Cout;
    int b = idx / (P * Cout);
    const float* ib   = in + (size_t)b * Cin * P;
    const float* wrow = wgt + (size_t)o * Cin;
    float acc = 0.f;
    for (int c = 0; c < Cin; ++c) {
        float v = ib[(size_t)c * P + p];
        if (MODE >= 1) v = fmaxf((v - mu[c]) * rs[c] * gamma[c] + beta[c], 0.f);
        if (MODE == 2) v += res[(size_t)b * Cin * P + (size_t)c * P + p];
        acc += wrow[c] * v;
    }
    if (outAdd) acc += outAdd[idx];
    out[idx] = acc;
}

// ---------------------------------------------------------------------------
// Per-channel training-mode BN statistics over (B, P): mu[c], rs[c]=rsqrt(var+eps)
// grid = C channels, block = 256
// ---------------------------------------------------------------------------
__global__ void bn_stats_kernel(const float* __restrict__ in, int B, int C, int P,
                                float* __restrict__ mu, float* __restrict__ rs) {
    int c   = blockIdx.x;
    int tid = threadIdx.x;
    int M   = B * P;
    float s = 0.f, s2 = 0.f;
    for (int i = tid; i < M; i += blockDim.x) {
        int b = i / P, p = i - b * P;
        float v = in[((size_t)b * C + c) * P + p];
        s += v;
        s2 += v * v;
    }
    __shared__ float sh1[256], sh2[256];
    sh1[tid] = s;
    sh2[tid] = s2;
    __syncthreads();
    for (int st = 128; st > 0; st >>= 1) {
        if (tid < st) { sh1[tid] += sh1[tid + st]; sh2[tid] += sh2[tid + st]; }
        __syncthreads();
    }
    if (tid == 0) {
        float m   = sh1[0] / (float)M;
        float var = sh2[0] / (float)M - m * m;
        mu[c] = m;
        rs[c] = rsqrtf(var + 1e-5f);
    }
}

// ---------------------------------------------------------------------------
// Involution apply: out[b, cOff+c, h, w] = sum_j wk[b, g*k*k+j, h, w] * in_pad[...]
// wk layout [B, groups*k*k, H, W]; out layout [B, Ctot, H, W]
// ---------------------------------------------------------------------------
__global__ void involution_kernel(const float* __restrict__ in, const float* __restrict__ wk,
                                  float* __restrict__ out,
                                  int B, int C, int H, int W, int k, int groups,
                                  int Ctot, int cOff) {
    int idx   = blockIdx.x * blockDim.x + threadIdx.x;
    int total = B * C * H * W;
    if (idx >= total) return;
    int w0 = idx % W;
    int h  = (idx / W) % H;
    int c  = (idx / (W * H)) % C;
    int b  = idx / (W * H * C);
    int Cg  = C / groups;
    int g   = c / Cg;
    int pad = (k - 1) >> 1;
    int HW  = H * W;
    const float* wkb = wk + ((size_t)b * groups + g) * (k * k) * HW + h * W + w0;
    const float* inb = in + ((size_t)b * C + c) * HW;
    float acc = 0.f;
    for (int dy = 0; dy < k; ++dy) {
        int hh = h + dy - pad;
        if ((unsigned)hh >= (unsigned)H) continue;
        for (int dx = 0; dx < k; ++dx) {
            int ww = w0 + dx - pad;
            if ((unsigned)ww >= (unsigned)W) continue;
            acc += wkb[(size_t)(dy * k + dx) * HW] * inb[hh * W + ww];
        }
    }
    out[((size_t)b * Ctot + cOff + c) * HW + h * W + w0] = acc;
}

// out = y * (loc + glb), elementwise
__global__ void gate_kernel(const float* __restrict__ y, const float* __restrict__ loc,
                            const float* __restrict__ glb, float* __restrict__ out, int n) {
    int i = blockIdx.x * blockDim.x + threadIdx.x;
    if (i < n) out[i] = y[i] * (loc[i] + glb[i]);
}

// ---------------------------------------------------------------------------
// sigma[b, c, d] = (1/N) sum_n X[b,c,n] * (X[b,d,n] - mean_d)     (padded CPADxCPAD)
// One workgroup per sample, (CPAD/16)^2 waves, WMMA f32 16x16x4.
// Dynamic LDS: A[CPAD*KPAD] + B[KPAD*CPAD] + mu[CPAD]
// ---------------------------------------------------------------------------
template <int CPAD>
__global__ void sigma_kernel(const float* __restrict__ in, float* __restrict__ sigma,
                             int C, int N, int KPAD) {
    const int b = blockIdx.x;
    extern __shared__ float lds[];
    float* A   = lds;                 // [CPAD][KPAD] : X^T (uncentered)
    float* Bm  = A + CPAD * KPAD;     // [KPAD][CPAD] : Xc (centered)
    float* muL = Bm + KPAD * CPAD;    // [CPAD]
    const int tid = threadIdx.x, nth = blockDim.x;
    const float* xb = in + (size_t)b * C * N;

    for (int c = tid; c < CPAD; c += nth) {
        float s = 0.f;
        if (c < C)
            for (int n = 0; n < N; ++n) s += xb[(size_t)c * N + n];
        muL[c] = s / (float)N;
    }
    __syncthreads();
    for (int i = tid; i < CPAD * KPAD; i += nth) {
        int c = i / KPAD, n = i % KPAD;
        A[i] = (c < C && n < N) ? xb[(size_t)c * N + n] : 0.f;
    }
    for (int i = tid; i < KPAD * CPAD; i += nth) {
        int n = i / CPAD, c = i % CPAD;
        Bm[i] = (c < C && n < N) ? (xb[(size_t)c * N + n] - muL[c]) : 0.f;
    }
    __syncthreads();

    const int lane = tid & 31, wave = tid >> 5;
    constexpr int tilesN = CPAD / 16;
    const int mi = wave / tilesN, ni = wave % tilesN;
    v8f acc = v8f_zero();
    acc = wmma_f32_mm(A, KPAD, Bm, CPAD, KPAD, mi, ni, lane, acc);
    wmma_store(sigma + (size_t)b * CPAD * CPAD, CPAD, mi, ni, lane, acc, 1.0f / (float)N);
}

// ---------------------------------------------------------------------------
// Newton-Schulz (6 iters), faithful to the reference:
//   T = 3I - Z@Y ;  Y = 0.5*Y (elementwise*) T ;  Z = (0.5*T)@Z
// then s = sqrt(tr(Y)), feat[c] = s * mean_rows(Y)[c].
// One workgroup per sample; Y/Z/T live in LDS; matmuls via WMMA f32 16x16x4.
// ---------------------------------------------------------------------------
template <int CPAD>
__global__ void ns_kernel(const float* __restrict__ sigma, float* __restrict__ feat,
                          int C, int featStride, int featOff, float invC) {
    constexpr int M = CPAD * CPAD;
    __shared__ float Y[M];
    __shared__ float T[M];
    __shared__ float Zb[2][M];
    __shared__ float trsh;
    const int b = blockIdx.x;
    const int tid = threadIdx.x, nth = blockDim.x;
    const float* sg = sigma + (size_t)b * M;

    if (tid == 0) {
        float t = 0.f;
        for (int i = 0; i < C; ++i) t += sg[i * CPAD + i];
        trsh = t;
    }
    __syncthreads();
    const float invtr = 1.0f / trsh;
    for (int i = tid; i < M; i += nth) {
        Y[i] = sg[i] * invtr;
        int r = i / CPAD, c = i % CPAD;
        Zb[0][i] = (r == c) ? 1.f : 0.f;
    }
    __syncthreads();

    const int lane = tid & 31, wave = tid >> 5;
    constexpr int tilesN = CPAD / 16;
    const int mi = wave / tilesN, ni = wave % tilesN;
    const int col = ni * 16 + (lane & 15);
    const int rbase = mi * 16 + (lane >> 4) * 8;

    int zp = 0;
    for (int it = 0; it < 6; ++it) {
        // T = 3I - Z@Y
        v8f acc = v8f_zero();
        acc = wmma_f32_mm(Zb[zp], CPAD, Y, CPAD, CPAD, mi, ni, lane, acc);
        for (int r2 = 0; r2 < 8; ++r2) {
            int row = rbase + r2;
            T[row * CPAD + col] = ((row == col) ? 3.f : 0.f) - acc[r2];
        }
        __syncthreads();
        // Znew = 0.5 * T @ Zold
        v8f acc2 = v8f_zero();
        acc2 = wmma_f32_mm(T, CPAD, Zb[zp], CPAD, CPAD, mi, ni, lane, acc2);
        for (int r2 = 0; r2 < 8; ++r2) {
            int row = rbase + r2;
            Zb[1 - zp][row * CPAD + col] = 0.5f * acc2[r2];
        }
        // Y = 0.5 * Y (elementwise) T   (T fixed after barrier; no wave reads Y now)
        for (int i = tid; i < M; i += nth) Y[i] = 0.5f * Y[i] * T[i];
        __syncthreads();
        zp ^= 1;
    }

    if (tid == 0) {
        float t = 0.f;
        for (int i = 0; i < C; ++i) t += Y[i * CPAD + i];
        trsh = sqrtf(t);
    }
    __syncthreads();
    const float s = trsh;
    for (int c = tid; c < C; c += nth) {
        float sum = 0.f;
        for (int r = 0; r < C; ++r) sum += Y[r * CPAD + c];
        feat[(size_t)b * featStride + featOff + c] = s * sum * invC;
    }
}

// ---------------------------------------------------------------------------
// Head: relu(bn(feat[46])) -> linear (16,46) -> softmax.  grid=B, block=64.
// ---------------------------------------------------------------------------
__global__ void head_kernel(const float* __restrict__ feat,
                            const float* __restrict__ mu, const float* __restrict__ rs,
                            const float* __restrict__ g, const float* __restrict__ bb,
                            const float* __restrict__ lw, const float* __restrict__ lb,
                            float* __restrict__ out) {
    const int b = blockIdx.x, tid = threadIdx.x;
    __shared__ float f[46];
    __shared__ float lg[16];
    for (int c = tid; c < 46; c += blockDim.x) {
        float v = feat[(size_t)b * 46 + c];
        v = (v - mu[c]) * rs[c] * g[c] + bb[c];
        f[c] = fmaxf(v, 0.f);
    }
    __syncthreads();
    if (tid < 16) {
        float acc = lb[tid];
        for (int c = 0; c < 46; ++c) acc += f[c] * lw[tid * 46 + c];
        lg[tid] = acc;
    }
    __syncthreads();
    if (tid == 0) {
        float mx = lg[0];
        for (int j = 1; j < 16; ++j) mx = fmaxf(mx, lg[j]);
        float e[16], s = 0.f;
        for (int j = 0; j < 16; ++j) { e[j] = expf(lg[j] - mx); s += e[j]; }
        float inv = 1.f / s;
        for (int j = 0; j < 16; ++j) out[(size_t)b * 16 + j] = e[j] * inv;
    }
}

// ---------------------------------------------------------------------------
// Host orchestration
// ---------------------------------------------------------------------------
static inline int cdiv(int a, int b) { return (a + b - 1) / b; }

extern "C" void kernel_launch(void* const* d_in, const int* in_sizes, int n_in,
                              void* d_out, int out_size, void* d_ws, size_t ws_size,
                              hipStream_t stream) {
    (void)in_sizes; (void)n_in; (void)out_size; (void)ws_size;
    int ai = 0;
    const float* x        = (const float*)d_in[ai++];
    const float* y        = (const float*)d_in[ai++];
    const float* blk_w1   = (const float*)d_in[ai++];
    const float* blk_ig   = (const float*)d_in[ai++];
    const float* blk_ib   = (const float*)d_in[ai++];
    const float* blk_w2_5 = (const float*)d_in[ai++];
    const float* blk_w2_7 = (const float*)d_in[ai++];
    const float* blk_w2_9 = (const float*)d_in[ai++];
    const float* blk_g    = (const float*)d_in[ai++];
    const float* blk_b    = (const float*)d_in[ai++];
    const float* blk_conv = (const float*)d_in[ai++];
    const float* ca_loc_w1 = (const float*)d_in[ai++];
    const float* ca_loc_g  = (const float*)d_in[ai++];
    const float* ca_loc_b  = (const float*)d_in[ai++];
    const float* ca_loc_w2 = (const float*)d_in[ai++];
    const float* ca_glb_w1 = (const float*)d_in[ai++];
    const float* ca_glb_g  = (const float*)d_in[ai++];
    const float* ca_glb_b  = (const float*)d_in[ai++];
    const float* ca_glb_w2 = (const float*)d_in[ai++];
    const float* ca_c3_w1  = (const float*)d_in[ai++];
    const float* ca_c3_g   = (const float*)d_in[ai++];
    const float* ca_c3_b   = (const float*)d_in[ai++];
    const float* ca_c3_w2  = (const float*)d_in[ai++];
    const float* ca_c4_w1  = (const float*)d_in[ai++];
    const float* ca_c4_g   = (const float*)d_in[ai++];
    const float* ca_c4_b   = (const float*)d_in[ai++];
    const float* ca_c4_w2  = (const float*)d_in[ai++];
    const float* ca_c5_w1  = (const float*)d_in[ai++];
    const float* ca_c5_g   = (const float*)d_in[ai++];
    const float* ca_c5_b   = (const float*)d_in[ai++];
    const float* ca_c5_w2  = (const float*)d_in[ai++];
    const float* ca_p1     = (const float*)d_in[ai++];
    const float* ca_p2     = (const float*)d_in[ai++];
    const float* ca_p3     = (const float*)d_in[ai++];
    const float* ca_bn1g   = (const float*)d_in[ai++];
    const float* ca_bn1b   = (const float*)d_in[ai++];
    const float* ca_bn2g   = (const float*)d_in[ai++];
    const float* ca_bn2b   = (const float*)d_in[ai++];
    const float* ca_bn3g   = (const float*)d_in[ai++];
    const float* ca_bn3b   = (const float*)d_in[ai++];
    const float* hbn_g     = (const float*)d_in[ai++];
    const float* hbn_b     = (const float*)d_in[ai++];
    const float* lin_w     = (const float*)d_in[ai++];
    const float* lin_b     = (const float*)d_in[ai++];
    float* out_p = (float*)d_out;

    const int B = 64, P1 = 225, P2 = 9;

    // -- bump allocator over d_ws --
    float* ws = (float*)d_ws;
    size_t off = 0;
    auto alloc = [&](size_t n) {
        float* p = ws + off;
        off += (n + 63) & ~(size_t)63;
        return p;
    };
    float* outA  = alloc((size_t)B * 30 * P1);
    float* outB  = alloc((size_t)B * 30 * P1);
    float* tbuf  = alloc((size_t)B * 102 * P1);   // shared t scratch (max 15*225 / 102*9)
    float* wkbuf = alloc((size_t)B * 162 * P1);   // shared wk scratch (max 162*225 / 1836*9)
    float* o90   = alloc((size_t)B * 90 * P1);
    float* sig1  = alloc((size_t)B * 32 * 32);
    float* loc   = alloc((size_t)B * 204 * P2);
    float* glb   = alloc((size_t)B * 204 * P2);
    float* o0    = alloc((size_t)B * 204 * P2);
    float* c3o   = alloc((size_t)B * 204 * P2);
    float* o1    = alloc((size_t)B * 102 * P2);
    float* c4o   = alloc((size_t)B * 102 * P2);
    float* o2    = alloc((size_t)B * 51 * P2);
    float* c5o   = alloc((size_t)B * 51 * P2);
    float* o3    = alloc((size_t)B * 16 * P2);
    float* sig2  = alloc((size_t)B * 16 * 16);
    float* feat  = alloc((size_t)B * 46);
    float* mu    = alloc(256);
    float* rs    = alloc(256);

    // ================= Branch 1: 3 IVoubottleneck blocks =================
    const float* cur = x;
    float* outbuf[2] = {outA, outB};
    const int ksz[3] = {5, 7, 9};
    const float* w2base[3]  = {blk_w2_5, blk_w2_7, blk_w2_9};
    const int    w2elems[3] = {50 * 15, 98 * 15, 162 * 15};

    for (int bi = 0; bi < 3; ++bi) {
        for (int ii = 0; ii < 3; ++ii) {
            const int k = ksz[ii], kkg = k * k * 2;
            const float* w1p = blk_w1 + (size_t)(bi * 3 + ii) * 15 * 30;
            const float* igp = blk_ig + (bi * 3 + ii) * 15;
            const float* ibp = blk_ib + (bi * 3 + ii) * 15;
            const float* w2p = w2base[ii] + (size_t)bi * w2elems[ii];
            conv1x1_kernel<0><<<cdiv(B * 15 * P1, 256), 256, 0, stream>>>(
                cur, w1p, tbuf, nullptr, nullptr, nullptr, nullptr, nullptr, nullptr,
                B, 30, 15, P1);
            bn_stats_kernel<<<15, 256, 0, stream>>>(tbuf, B, 15, P1, mu, rs);
            conv1x1_kernel<1><<<cdiv(B * kkg * P1, 256), 256, 0, stream>>>(
                tbuf, w2p, wkbuf, mu, rs, igp, ibp, nullptr, nullptr,
                B, 15, kkg, P1);
            involution_kernel<<<cdiv(B * 30 * P1, 256), 256, 0, stream>>>(
                cur, wkbuf, o90, B, 30, 15, 15, k, 2, 90, ii * 30);
        }
        bn_stats_kernel<<<90, 256, 0, stream>>>(o90, B, 90, P1, mu, rs);
        float* nxt = outbuf[bi & 1];
        conv1x1_kernel<1><<<cdiv(B * 30 * P1, 256), 256, 0, stream>>>(
            o90, blk_conv + (size_t)bi * 30 * 90, nxt, mu, rs,
            blk_g + bi * 90, blk_b + bi * 90, nullptr, cur, B, 90, 30, P1);
        cur = nxt;
    }
    // cov_pool on [B,30,225] : CPAD=32, K padded 225->228
    {
        const size_t ldsB = (size_t)(32 * 228 + 228 * 32 + 32) * sizeof(float);
        sigma_kernel<32><<<B, 128, ldsB, stream>>>(cur, sig1, 30, 225, 228);
        ns_kernel<32><<<B, 128, 0, stream>>>(sig1, feat, 30, 46, 0, 1.f / 30.f);
    }

    // ================= Branch 2: cross_attention =================
    auto involution_full = [&](const float* in, int C, int Cmid,
                               const float* w1p, const float* gp, const float* bp,
                               const float* w2p, int groups, float* outp) {
        const int kkg = 9 * groups;
        conv1x1_kernel<0><<<cdiv(B * Cmid * P2, 256), 256, 0, stream>>>(
            in, w1p, tbuf, nullptr, nullptr, nullptr, nullptr, nullptr, nullptr,
            B, C, Cmid, P2);
        bn_stats_kernel<<<Cmid, 256, 0, stream>>>(tbuf, B, Cmid, P2, mu, rs);
        conv1x1_kernel<1><<<cdiv(B * kkg * P2, 256), 256, 0, stream>>>(
            tbuf, w2p, wkbuf, mu, rs, gp, bp, nullptr, nullptr, B, Cmid, kkg, P2);
        involution_kernel<<<cdiv(B * C * P2, 256), 256, 0, stream>>>(
            in, wkbuf, outp, B, C, 3, 3, 3, groups, C, 0);
    };

    involution_full(y, 204, 102, ca_loc_w1, ca_loc_g, ca_loc_b, ca_loc_w2, 204, loc);
    involution_full(y, 204, 102, ca_glb_w1, ca_glb_g, ca_glb_b, ca_glb_w2, 1, glb);
    gate_kernel<<<cdiv(B * 204 * P2, 256), 256, 0, stream>>>(y, loc, glb, o0, B * 204 * P2);

    involution_full(o0, 204, 102, ca_c3_w1, ca_c3_g, ca_c3_b, ca_c3_w2, 1, c3o);
    bn_stats_kernel<<<204, 256, 0, stream>>>(c3o, B, 204, P2, mu, rs);
    // o1 = p1 @ (relu(bn(c3o)) + o0)
    conv1x1_kernel<2><<<cdiv(B * 102 * P2, 256), 256, 0, stream>>>(
        c3o, ca_p1, o1, mu, rs, ca_bn1g, ca_bn1b, o0, nullptr, B, 204, 102, P2);

    involution_full(o1, 102, 51, ca_c4_w1, ca_c4_g, ca_c4_b, ca_c4_w2, 1, c4o);
    bn_stats_kernel<<<102, 256, 0, stream>>>(c4o, B, 102, P2, mu, rs);
    // o2 = p2 @ relu(bn(c4o))   (no residual, matches source)
    conv1x1_kernel<1><<<cdiv(B * 51 * P2, 256), 256, 0, stream>>>(
        c4o, ca_p2, o2, mu, rs, ca_bn2g, ca_bn2b, nullptr, nullptr, B, 102, 51, P2);

    involution_full(o2, 51, 25, ca_c5_w1, ca_c5_g, ca_c5_b, ca_c5_w2, 1, c5o);
    bn_stats_kernel<<<51, 256, 0, stream>>>(c5o, B, 51, P2, mu, rs);
    // o3 = p3 @ (relu(bn(c5o)) + o2)
    conv1x1_kernel<2><<<cdiv(B * 16 * P2, 256), 256, 0, stream>>>(
        c5o, ca_p3, o3, mu, rs, ca_bn3g, ca_bn3b, o2, nullptr, B, 51, 16, P2);

    // cov_pool on [B,16,9] : CPAD=16, K padded 9->12
    {
        const size_t ldsB = (size_t)(16 * 12 + 12 * 16 + 16) * sizeof(float);
        sigma_kernel<16><<<B, 32, ldsB, stream>>>(o3, sig2, 16, 9, 12);
        ns_kernel<16><<<B, 32, 0, stream>>>(sig2, feat, 16, 46, 30, 1.f / 16.f);
    }

    // ================= Head =================
    bn_stats_kernel<<<46, 256, 0, stream>>>(feat, B, 46, 1, mu, rs);
    head_kernel<<<B, 64, 0, stream>>>(feat, mu, rs, hbn_g, hbn_b, lin_w, lin_b, out_p);
}